// RewardModel_63556926046695
// MI455X (gfx1250) — compile-verified
//
#include <hip/hip_runtime.h>
#include <math.h>

typedef __attribute__((ext_vector_type(2))) float v2f;
typedef __attribute__((ext_vector_type(8))) float v8f;

#define BSZ     16
#define SEQ     2048
#define DIM     2048
#define PAD_ID  1

// ---------------------------------------------------------------------------
// Kernel 1: rewards[b*SEQ + s] = dot(hidden[b,s,:], w)  via V_WMMA_F32_16X16X4_F32
// One wave handles 16 consecutive (b,s) rows. M = seq rows, K = 4 d-values,
// B broadcasts w across all 16 N columns (every column of D equals the dot).
// ---------------------------------------------------------------------------
__global__ __launch_bounds__(256) void gemv_wmma(const float* __restrict__ h,
                                                 const float* __restrict__ w,
                                                 float* __restrict__ rewards) {
    const int lane = threadIdx.x & 31;
    const int wave = threadIdx.x >> 5;
    const int tile = blockIdx.x * 8 + wave;          // 4096 tiles of 16 rows
    const int m    = lane & 15;                      // row within tile
    const int koff = (lane >> 4) << 1;               // 0 (lanes 0-15) or 2 (lanes 16-31)

    const float* hrow = h + (size_t)(tile * 16 + m) * DIM + koff;
    const float* wp   = w + koff;

    v8f acc0 = {0.f,0.f,0.f,0.f,0.f,0.f,0.f,0.f};
    v8f acc1 = acc0, acc2 = acc0, acc3 = acc0;

    for (int d = 0; d < DIM; d += 16) {
        // prefetch ahead along the stream
        __builtin_prefetch(hrow + d + 512, 0, 0);

        v2f a0 = *(const v2f*)(hrow + d);
        v2f b0 = *(const v2f*)(wp   + d);
        v2f a1 = *(const v2f*)(hrow + d + 4);
        v2f b1 = *(const v2f*)(wp   + d + 4);
        v2f a2 = *(const v2f*)(hrow + d + 8);
        v2f b2 = *(const v2f*)(wp   + d + 8);
        v2f a3 = *(const v2f*)(hrow + d + 12);
        v2f b3 = *(const v2f*)(wp   + d + 12);

        acc0 = __builtin_amdgcn_wmma_f32_16x16x4_f32(false, a0, false, b0,
                                                     (short)0, acc0, false, false);
        acc1 = __builtin_amdgcn_wmma_f32_16x16x4_f32(false, a1, false, b1,
                                                     (short)0, acc1, false, false);
        acc2 = __builtin_amdgcn_wmma_f32_16x16x4_f32(false, a2, false, b2,
                                                     (short)0, acc2, false, false);
        acc3 = __builtin_amdgcn_wmma_f32_16x16x4_f32(false, a3, false, b3,
                                                     (short)0, acc3, false, false);
    }

    v8f acc = (acc0 + acc1) + (acc2 + acc3);

    // D layout: lane L (N=L%16), VGPR r -> M = r + 8*(L/16). All N columns equal.
    if (m == 0) {
        float* out = rewards + tile * 16 + ((lane >> 4) << 3);
        #pragma unroll
        for (int r = 0; r < 8; ++r) out[r] = acc[r];
    }
}

// ---------------------------------------------------------------------------
// Kernel 2: per-pair index logic + span softplus sum. One block per pair b.
// ---------------------------------------------------------------------------
__global__ __launch_bounds__(256) void epilogue(const int* __restrict__ ids,
                                                const float* __restrict__ rewards,
                                                float* __restrict__ out,
                                                float* __restrict__ loss_part) {
    const int b = blockIdx.x;                          // 0..15
    const int* cho = ids + (size_t)b * SEQ;
    const int* rej = ids + (size_t)(b + BSZ) * SEQ;
    const float* cr = rewards + (size_t)b * SEQ;
    const float* rr = rewards + (size_t)(b + BSZ) * SEQ;

    __shared__ int   s_fc, s_fr, s_sc, s_sr, s_dv;
    __shared__ float red [256];
    __shared__ int   redi[256];

    if (threadIdx.x == 0) { s_fc = SEQ; s_fr = SEQ; s_sc = SEQ; s_sr = SEQ; s_dv = SEQ; }
    __syncthreads();

    // pass 1: first pad index (chosen/rejected) and first divergence index
    int fc = SEQ, fr = SEQ, dv = SEQ;
    for (int i = threadIdx.x; i < SEQ; i += 256) {
        int c = cho[i], r = rej[i];
        if (c == PAD_ID && i < fc) fc = i;
        if (r == PAD_ID && i < fr) fr = i;
        if (c != r      && i < dv) dv = i;
    }
    atomicMin(&s_fc, fc); atomicMin(&s_fr, fr); atomicMin(&s_dv, dv);
    __syncthreads();
    const int firstc = s_fc, firstr = s_fr;

    // pass 2: second pad index = min i > first with pad (== _kth_index(mask,1,S))
    int sc = SEQ, sr = SEQ;
    for (int i = threadIdx.x; i < SEQ; i += 256) {
        if (i > firstc && cho[i] == PAD_ID && i < sc) sc = i;
        if (i > firstr && rej[i] == PAD_ID && i < sr) sr = i;
    }
    atomicMin(&s_sc, sc); atomicMin(&s_sr, sr);
    __syncthreads();

    const int  c_ind   = s_sc;                 // SEQ if <2 pads
    const int  r_raw   = s_sr;
    const bool any_div = (s_dv < SEQ);
    const int  div_ind = any_div ? s_dv : (SEQ - 1);
    const int  r_ind   = any_div ? r_raw : c_ind;
    const int  end_ind = any_div ? max(c_ind, r_ind) : SEQ;

    // pass 3: sum softplus(rej - cho) over [div_ind, end_ind)
    float acc = 0.f; int cnt = 0;
    for (int i = threadIdx.x; i < SEQ; i += 256) {
        if (i >= div_ind && i < end_ind) {
            float x = rr[i] - cr[i];
            acc += fmaxf(x, 0.f) + log1pf(expf(-fabsf(x)));   // stable softplus
            cnt += 1;
        }
    }
    red[threadIdx.x] = acc; redi[threadIdx.x] = cnt;
    __syncthreads();
    for (int s = 128; s > 0; s >>= 1) {
        if ((int)threadIdx.x < s) {
            red [threadIdx.x] += red [threadIdx.x + s];
            redi[threadIdx.x] += redi[threadIdx.x + s];
        }
        __syncthreads();
    }

    if (threadIdx.x == 0) {
        float denom = (float)max(redi[0], 1);
        loss_part[b] = red[0] / denom;
        int ci = c_ind - 1; if (ci < 0) ci = 0; if (ci > SEQ - 1) ci = SEQ - 1;
        int ri = r_ind - 1; if (ri < 0) ri = 0; if (ri > SEQ - 1) ri = SEQ - 1;
        out[1 + b]       = cr[ci];             // chosen_mean_scores
        out[1 + BSZ + b] = rr[ri];             // rejected_mean_scores
    }
}

// ---------------------------------------------------------------------------
// Kernel 3: mean of per-pair losses
// ---------------------------------------------------------------------------
__global__ void finalize(const float* __restrict__ loss_part, float* __restrict__ out) {
    if (threadIdx.x == 0) {
        float s = 0.f;
        #pragma unroll
        for (int i = 0; i < BSZ; ++i) s += loss_part[i];
        out[0] = s / (float)BSZ;
    }
}

extern "C" void kernel_launch(void* const* d_in, const int* in_sizes, int n_in,
                              void* d_out, int out_size, void* d_ws, size_t ws_size,
                              hipStream_t stream) {
    const float* h   = (const float*)d_in[0];   // (32, 2048, 2048) f32
    const float* w   = (const float*)d_in[1];   // (2048,)          f32
    const int*   ids = (const int*)  d_in[2];   // (32, 2048)       i32
    float* out = (float*)d_out;                 // 33 floats

    float* rewards   = (float*)d_ws;            // 32*2048 floats
    float* loss_part = rewards + 2 * BSZ * SEQ; // 16 floats

    // 32*2048 rows / 16 rows-per-wave / 8 waves-per-block = 512 blocks
    gemv_wmma<<<512, 256, 0, stream>>>(h, w, rewards);
    epilogue <<<BSZ, 256, 0, stream>>>(ids, rewards, out, loss_part);
    finalize <<<1, 32, 0, stream>>>(loss_part, out);
}